// MoERouter_12043088298372
// MI455X (gfx1250) — compile-verified
//
#include <hip/hip_runtime.h>

typedef __attribute__((ext_vector_type(2))) float v2f;
typedef __attribute__((ext_vector_type(4))) float v4f;
typedef __attribute__((ext_vector_type(8))) float v8f;

// Problem constants (fixed by the reference)
#define NTOK   4096          // B*S = 2*2048
#define DIM    1024
#define NEXP   8
#define TOPK   2
#define CAP    1280          // floor(2*1.25*4096/8), already even
#define NASSIGN (TOPK*NTOK)  // 8192

// d_out layout (floats, concatenated in return order)
#define OFF_EW 0ULL                                  // expert_weights [4096,8,1280]
#define OFF_EM (41943040ULL)                         // expert_mask    [4096,8,1280]
#define OFF_EB (83886080ULL)                         // expert_batches [8,1280,1024]
#define OFF_LG (94371840ULL)                         // router_logits  [2,2048,8]
#define N_ZERO_F4 (OFF_LG / 4ULL)                    // 23,592,960 float4s to zero

// ---------------------------------------------------------------------------
// Kernel 1: router logits GEMM via V_WMMA_F32_16X16X4_F32.
// One wave per 16-token M-tile; N=16 (experts 8..15 zeroed); K in steps of 4.
// ---------------------------------------------------------------------------
__global__ void moe_router_gemm(const float* __restrict__ x,
                                const float* __restrict__ wg,
                                float* __restrict__ logits) {
  const int lane = threadIdx.x & 31;
  const int wave = threadIdx.x >> 5;
  const int tile = blockIdx.x * (blockDim.x >> 5) + wave;   // 256 tiles total
  const int half = lane >> 4;        // 0: K pair {0,1}, 1: K pair {2,3}
  const int row  = lane & 15;        // A: token-in-tile; B: expert column

  const float* ap = x + (size_t)(tile * 16 + row) * DIM + half * 2;
  const int brow = (row < NEXP) ? row : 0;                  // clamp to stay in-bounds
  const float bscale = (row < NEXP) ? 1.0f : 0.0f;          // zero pad experts 8..15
  const float* bp = wg + (size_t)brow * DIM + half * 2;

  v8f acc = {};
  #pragma unroll 4
  for (int k0 = 0; k0 < DIM; k0 += 4) {
    v2f a = *(const v2f*)(ap + k0);
    v2f b = *(const v2f*)(bp + k0);
    b *= bscale;                      // keeps EXEC all-1s through the WMMA
    acc = __builtin_amdgcn_wmma_f32_16x16x4_f32(
        /*neg_a=*/false, a, /*neg_b=*/false, b,
        /*c_mod=*/(short)0, acc, /*reuse_a=*/false, /*reuse_b=*/false);
  }

  // C/D layout: VGPR r -> M = r + 8*half, N = lane%16. Only experts 0..7 exist.
  if (row < NEXP) {
    #pragma unroll
    for (int r = 0; r < 8; ++r) {
      const int m = tile * 16 + r + half * 8;
      logits[(size_t)m * NEXP + row] = acc[r];
    }
  }
}

// ---------------------------------------------------------------------------
// Kernel 2: per-token top-2 (jax.lax.top_k tie rule: lower index wins) and
// softmax over the two kept logits.
// ---------------------------------------------------------------------------
__global__ void moe_topk(const float* __restrict__ logits,
                         int* __restrict__ topi, float* __restrict__ topp) {
  const int n = blockIdx.x * blockDim.x + threadIdx.x;
  if (n >= NTOK) return;
  float v[NEXP];
  #pragma unroll
  for (int e = 0; e < NEXP; ++e) v[e] = logits[(size_t)n * NEXP + e];

  int i0 = 0; float v0 = v[0];
  #pragma unroll
  for (int e = 1; e < NEXP; ++e) { if (v[e] > v0) { v0 = v[e]; i0 = e; } }
  int i1 = -1; float v1 = -__builtin_inff();
  #pragma unroll
  for (int e = 0; e < NEXP; ++e) { if (e != i0 && v[e] > v1) { v1 = v[e]; i1 = e; } }

  const float e1  = __expf(v1 - v0);          // exp shifted by max
  const float inv = 1.0f / (1.0f + e1);
  topi[n * 2 + 0] = i0;  topp[n * 2 + 0] = inv;
  topi[n * 2 + 1] = i1;  topp[n * 2 + 1] = e1 * inv;
}

// ---------------------------------------------------------------------------
// Kernel 3: deterministic per-expert ranking in k-major cumsum order.
// Single wave32; ballot multi-split; all 8 running bases live in registers
// (replicated per lane) — no LDS, no barriers.
// ---------------------------------------------------------------------------
__global__ void moe_rank(const int* __restrict__ topi, int* __restrict__ slots) {
  const int lane = threadIdx.x;
  int base[NEXP];
  #pragma unroll
  for (int e = 0; e < NEXP; ++e) base[e] = 0;

  const unsigned ltmask = (lane == 0) ? 0u : (0xffffffffu >> (32 - lane));

  for (int it = 0; it < NASSIGN / 32; ++it) {
    const int i = it * 32 + lane;            // k-major order: i = k*N + n
    const int k = i >> 12;                   // / 4096
    const int n = i & (NTOK - 1);
    const int e = topi[n * 2 + k];

    unsigned bal[NEXP];
    #pragma unroll
    for (int ee = 0; ee < NEXP; ++ee)
      bal[ee] = __builtin_amdgcn_ballot_w32(e == ee);

    int myBase = 0; unsigned myBal = 0;
    #pragma unroll
    for (int ee = 0; ee < NEXP; ++ee)
      if (e == ee) { myBase = base[ee]; myBal = bal[ee]; }

    const int rank = myBase + __popc(myBal & ltmask);
    slots[i] = (rank < CAP) ? rank : -1;

    #pragma unroll
    for (int ee = 0; ee < NEXP; ++ee) base[ee] += __popc(bal[ee]);
  }
}

// ---------------------------------------------------------------------------
// Kernel 0: zero-fill expert_weights + expert_mask + expert_batches (378 MB).
// This is the roofline term (~16 us at 23.3 TB/s): NT 128-bit stores.
// ---------------------------------------------------------------------------
__global__ void moe_zero(float* __restrict__ p, long long n4) {
  long long i = (long long)blockIdx.x * blockDim.x + threadIdx.x;
  const long long stride = (long long)gridDim.x * blockDim.x;
  const v4f z = {};
  for (; i < n4; i += stride)
    __builtin_nontemporal_store(z, ((v4f*)p) + i);
}

// ---------------------------------------------------------------------------
// Kernel 4: scatter gate weight / mask into the kept (n, e, slot) cells.
// ---------------------------------------------------------------------------
__global__ void moe_scatter(const int* __restrict__ topi,
                            const float* __restrict__ topp,
                            const int* __restrict__ slots,
                            float* __restrict__ ew, float* __restrict__ em) {
  const int i = blockIdx.x * blockDim.x + threadIdx.x;
  if (i >= NASSIGN) return;
  const int slot = slots[i];
  if (slot < 0) return;
  const int k = i >> 12, n = i & (NTOK - 1);
  const int e = topi[n * 2 + k];
  const float p = topp[n * 2 + k];
  if (!(p > 0.0f)) return;                   // reference: mask = weights > 0
  const size_t idx = (size_t)n * (NEXP * CAP) + (size_t)e * CAP + slot;
  ew[idx] = p;
  em[idx] = 1.0f;
}

// ---------------------------------------------------------------------------
// Kernel 5: gather token rows into expert batches (one block per assignment;
// 256 threads x float4 = 1024 floats). Each (e,slot) is written by exactly
// one assignment since ranks are unique per expert.
// ---------------------------------------------------------------------------
__global__ void moe_gather(const float* __restrict__ x,
                           const int* __restrict__ topi,
                           const float* __restrict__ topp,
                           const int* __restrict__ slots,
                           float* __restrict__ eb) {
  const int i = blockIdx.x;
  const int slot = slots[i];
  if (slot < 0) return;
  const int k = i >> 12, n = i & (NTOK - 1);
  if (!(topp[n * 2 + k] > 0.0f)) return;
  const int e = topi[n * 2 + k];
  const v4f* __restrict__ src = (const v4f*)(x + (size_t)n * DIM);
  v4f* __restrict__ dst = (v4f*)(eb + ((size_t)e * CAP + slot) * DIM);
  __builtin_nontemporal_store(src[threadIdx.x], dst + threadIdx.x);
}

// ---------------------------------------------------------------------------
extern "C" void kernel_launch(void* const* d_in, const int* in_sizes, int n_in,
                              void* d_out, int out_size, void* d_ws, size_t ws_size,
                              hipStream_t stream) {
  const float* x  = (const float*)d_in[0];   // [2,2048,1024] f32
  const float* wg = (const float*)d_in[1];   // [8,1024] f32

  float* out = (float*)d_out;
  float* ew = out + OFF_EW;
  float* em = out + OFF_EM;
  float* eb = out + OFF_EB;
  float* lg = out + OFF_LG;

  int*   topi  = (int*)d_ws;                                   // [4096,2]
  float* topp  = (float*)((char*)d_ws + NASSIGN * sizeof(int));// [4096,2]
  int*   slots = (int*)((char*)d_ws + 2 * NASSIGN * sizeof(int)); // [8192]

  // Zero-fill the big sparse outputs (bandwidth-dominant step).
  moe_zero<<<2048, 256, 0, stream>>>(out, (long long)N_ZERO_F4);
  // Router logits GEMM: 256 M-tiles = 32 blocks x 8 waves.
  moe_router_gemm<<<32, 256, 0, stream>>>(x, wg, lg);
  // Top-2 gating.
  moe_topk<<<NTOK / 256, 256, 0, stream>>>(lg, topi, topp);
  // Deterministic k-major capacity ranking (single wave).
  moe_rank<<<1, 32, 0, stream>>>(topi, slots);
  // Sparse scatter of weights/mask.
  moe_scatter<<<NASSIGN / 256, 256, 0, stream>>>(topi, topp, slots, ew, em);
  // Gather token rows into expert batches.
  moe_gather<<<NASSIGN, 256, 0, stream>>>(x, topi, topp, slots, eb);
}